// RejectionSampler_44040594653459
// MI455X (gfx1250) — compile-verified
//
#include <hip/hip_runtime.h>
#include <hip/hip_bf16.h>
#include <stdint.h>

// Rejection sampler (speculative decoding), B=64, K=8, V=128000.
// Memory-bound: ~64MB fp32 streamed once -> ~3us floor at 23.3 TB/s.
// No matrix math -> WMMA not applicable; use CDNA5 async global->LDS
// (ASYNCcnt) streaming + wave32 ballot + monotone-CDF crossing trick.

#define CHUNK    256               // floats per chunk (1 KB)
#define WAVES1   8
#define TPB1     (WAVES1 * 32)
#define SPLIT    16                // blocks per batch row in pass 1
#define TPB2     256
#define MAXCHUNK 1024              // supports V up to 262144

#if defined(__AMDGCN__) && __has_builtin(__builtin_amdgcn_global_load_async_to_lds_b128)
#define HAS_ASYNC_LDS 1
#else
#define HAS_ASYNC_LDS 0
#endif

#define GLOBAL_AS __attribute__((address_space(1)))
#define LDS_AS    __attribute__((address_space(3)))

typedef int v4i __attribute__((vector_size(16)));   // builtin's expected pointee

template <int N>
__device__ __forceinline__ void wait_async() {
#if defined(__AMDGCN__)
#if __has_builtin(__builtin_amdgcn_s_wait_asynccnt)
  __builtin_amdgcn_s_wait_asynccnt(N);
  asm volatile("" ::: "memory");
#else
  asm volatile("s_wait_asynccnt %0" : : "i"(N) : "memory");
#endif
#endif
}

__device__ __forceinline__ float wave_reduce_add(float x) {
  for (int off = 16; off > 0; off >>= 1) x += __shfl_down(x, off, 32);
  return x;
}

// Must be executed by all 32 lanes of wave 0. Returns num_accepted (<= K).
__device__ __forceinline__ int compute_na(int b, int K, int V,
                                          const int* __restrict__ dtok,
                                          const float* __restrict__ dp,
                                          const float* __restrict__ op,
                                          const float* __restrict__ uni) {
  const int lane = threadIdx.x & 31;
  int acc = 0;
  if (lane < K) {
    int tok  = dtok[b * K + lane];
    float pd = dp[((size_t)b * K + lane) * (size_t)V + tok];
    float po = op[((size_t)b * (K + 1) + lane) * (size_t)V + tok];
    float ap = fminf(1.0f, po / pd);
    acc = (uni[b * K + lane] < ap) ? 1 : 0;
  }
  unsigned mask = (unsigned)__ballot(acc);   // wave32: bits >= K are zero
  return __ffs(~mask) - 1;                   // count of leading accepted
}

// ---------------- Kernel 1: per-chunk sums of the final distribution -------
// grid = (B, SPLIT), block = 256 (8 waves). Each wave streams 1KB chunks.
__global__ void __launch_bounds__(TPB1)
rs_chunk_sums(const int* __restrict__ dtok, const float* __restrict__ dp,
              const float* __restrict__ op, const float* __restrict__ uni,
              float* __restrict__ ws_cs, int B, int K, int V, int nchunk) {
  const int b = blockIdx.x;
  const int s = blockIdx.y;
  __shared__ int s_na;
  if (threadIdx.x < 32) {
    int na = compute_na(b, K, V, dtok, dp, op, uni);
    if (threadIdx.x == 0) s_na = na;
  }
  __syncthreads();
  const int  na      = s_na;
  const bool all_acc = (na == K);
  const int  r       = min(na, K - 1);
  const float* __restrict__ orow =
      op + ((size_t)b * (K + 1) + (all_acc ? K : r)) * (size_t)V;
  const float* __restrict__ drow = dp + ((size_t)b * K + r) * (size_t)V;

  const int cps = (nchunk + (int)gridDim.y - 1) / (int)gridDim.y;
  const int c0  = s * cps;
  const int c1  = min(nchunk, c0 + cps);
  const int w    = threadIdx.x >> 5;
  const int lane = threadIdx.x & 31;

#if HAS_ASYNC_LDS
  // Double-buffered async global->LDS staging (per-wave private buffers).
  __shared__ __align__(16) float stage[WAVES1][2][2 * CHUNK];  // 32 KB

  auto issue = [&](int j, int p) {
    const float* go = orow + (size_t)j * CHUNK + lane * 4;
    float*       ds = &stage[w][p][lane * 4];
    __builtin_amdgcn_global_load_async_to_lds_b128(
        (GLOBAL_AS v4i*)(GLOBAL_AS void*)(go),
        (LDS_AS v4i*)(LDS_AS void*)(ds), 0, 0);
    __builtin_amdgcn_global_load_async_to_lds_b128(
        (GLOBAL_AS v4i*)(GLOBAL_AS void*)(go + 128),
        (LDS_AS v4i*)(LDS_AS void*)(ds + 128), 0, 0);
    if (!all_acc) {
      const float* gd = drow + (size_t)j * CHUNK + lane * 4;
      __builtin_amdgcn_global_load_async_to_lds_b128(
          (GLOBAL_AS v4i*)(GLOBAL_AS void*)(gd),
          (LDS_AS v4i*)(LDS_AS void*)(ds + 256), 0, 0);
      __builtin_amdgcn_global_load_async_to_lds_b128(
          (GLOBAL_AS v4i*)(GLOBAL_AS void*)(gd + 128),
          (LDS_AS v4i*)(LDS_AS void*)(ds + 384), 0, 0);
    }
  };

  int nloc = 0;
  for (int j = c0 + w; j < c1; j += WAVES1) nloc++;
  if (nloc > 0) issue(c0 + w, 0);
  for (int i = 0; i < nloc; ++i) {
    const int  j    = c0 + w + i * WAVES1;
    const bool more = (i + 1 < nloc);
    if (more) issue(j + WAVES1, (i + 1) & 1);
    if (more) {
      if (all_acc) wait_async<2>(); else wait_async<4>();
    } else {
      wait_async<0>();
    }
    const int p = i & 1;
    const float4* sv = (const float4*)&stage[w][p][0];
    float4 a0 = sv[lane], a1 = sv[32 + lane];
    float sum;
    if (all_acc) {
      sum = a0.x + a0.y + a0.z + a0.w + a1.x + a1.y + a1.z + a1.w;
    } else {
      float4 b0 = sv[64 + lane], b1 = sv[96 + lane];
      sum = fmaxf(a0.x - b0.x, 0.f) + fmaxf(a0.y - b0.y, 0.f) +
            fmaxf(a0.z - b0.z, 0.f) + fmaxf(a0.w - b0.w, 0.f) +
            fmaxf(a1.x - b1.x, 0.f) + fmaxf(a1.y - b1.y, 0.f) +
            fmaxf(a1.z - b1.z, 0.f) + fmaxf(a1.w - b1.w, 0.f);
    }
    sum = wave_reduce_add(sum);
    if (lane == 0) ws_cs[(size_t)b * nchunk + j] = sum;
  }
#else
  // Fallback: direct float4 loads + global_prefetch of the next chunk.
  for (int j = c0 + w; j < c1; j += WAVES1) {
    const float4* go = (const float4*)(orow + (size_t)j * CHUNK);
    if (j + WAVES1 < c1)
      __builtin_prefetch(orow + (size_t)(j + WAVES1) * CHUNK + lane * 4, 0, 0);
    float4 a0 = go[lane], a1 = go[32 + lane];
    float sum;
    if (all_acc) {
      sum = a0.x + a0.y + a0.z + a0.w + a1.x + a1.y + a1.z + a1.w;
    } else {
      const float4* gd = (const float4*)(drow + (size_t)j * CHUNK);
      if (j + WAVES1 < c1)
        __builtin_prefetch(drow + (size_t)(j + WAVES1) * CHUNK + lane * 4, 0, 0);
      float4 b0 = gd[lane], b1 = gd[32 + lane];
      sum = fmaxf(a0.x - b0.x, 0.f) + fmaxf(a0.y - b0.y, 0.f) +
            fmaxf(a0.z - b0.z, 0.f) + fmaxf(a0.w - b0.w, 0.f) +
            fmaxf(a1.x - b1.x, 0.f) + fmaxf(a1.y - b1.y, 0.f) +
            fmaxf(a1.z - b1.z, 0.f) + fmaxf(a1.w - b1.w, 0.f);
    }
    sum = wave_reduce_add(sum);
    if (lane == 0) ws_cs[(size_t)b * nchunk + j] = sum;
  }
#endif
}

// ---------------- Kernel 2: scan chunk sums, locate CDF crossing, emit -----
// grid = B, block = 256.
__global__ void __launch_bounds__(TPB2)
rs_finalize(const int* __restrict__ dtok, const float* __restrict__ dp,
            const float* __restrict__ op, const float* __restrict__ uni,
            const float* __restrict__ su, const float* __restrict__ ws_cs,
            int* __restrict__ out, int B, int K, int V, int nchunk) {
  const int b = blockIdx.x;
  __shared__ int   s_na;
  __shared__ int   s_cnt;
  __shared__ float s_tot;
  __shared__ float s_cs[MAXCHUNK];
  __shared__ float s_ex[MAXCHUNK];

  if (threadIdx.x < 32) {
    int na = compute_na(b, K, V, dtok, dp, op, uni);
    if (threadIdx.x == 0) { s_na = na; s_cnt = 0; }
  }
  for (int j = threadIdx.x; j < nchunk; j += blockDim.x)
    s_cs[j] = ws_cs[(size_t)b * nchunk + j];
  __syncthreads();

  const int  na      = s_na;
  const bool all_acc = (na == K);
  const int  r       = min(na, K - 1);
  const float* __restrict__ orow =
      op + ((size_t)b * (K + 1) + (all_acc ? K : r)) * (size_t)V;
  const float* __restrict__ drow = dp + ((size_t)b * K + r) * (size_t)V;

  // Wave 0: exclusive scan of chunk sums (wave32 Hillis-Steele + carry).
  if (threadIdx.x < 32) {
    const int lane = threadIdx.x;
    float carry = 0.f;
    for (int base = 0; base < nchunk; base += 32) {
      int j   = base + lane;
      float x = (j < nchunk) ? s_cs[j] : 0.f;
      float sc = x;
      for (int off = 1; off < 32; off <<= 1) {
        float t = __shfl_up(sc, off, 32);
        if (lane >= off) sc += t;
      }
      if (j < nchunk) s_ex[j] = carry + (sc - x);  // exclusive prefix
      carry += __shfl(sc, 31, 32);
    }
    if (lane == 0) s_tot = carry;
  }
  __syncthreads();

  const float S = s_tot;
  const float u = su[b];
  const float T = all_acc ? u : u * S;  // count prefix(raw) < u*S == cdf < u

  int cnt = 0;
  for (int j = threadIdx.x; j < nchunk; j += blockDim.x) {
    const float Bj = s_ex[j];
    const float Ej = Bj + s_cs[j];
    if (Ej < T) {
      cnt += CHUNK;                    // monotone: whole chunk below target
    } else if (Bj < T) {
      float run = Bj;                  // crossing chunk: element rescan (L2)
      const size_t base = (size_t)j * CHUNK;
      for (int e = 0; e < CHUNK; ++e) {
        float v = orow[base + e];
        if (!all_acc) v = fmaxf(v - drow[base + e], 0.f);
        run += v;
        cnt += (run < T) ? 1 : 0;
      }
    }
  }
  atomicAdd(&s_cnt, cnt);
  __syncthreads();

  const int new_token = min(s_cnt, V - 1);
  const int t = threadIdx.x;
  if (t <= K) {
    int val;
    if (t < na)       val = dtok[b * K + t];
    else if (t == na) val = new_token;
    else              val = -1;
    out[b * (K + 1) + t] = val;
  }
  if (t == 0) out[B * (K + 1) + b] = na;
}

extern "C" void kernel_launch(void* const* d_in, const int* in_sizes, int n_in,
                              void* d_out, int out_size, void* d_ws, size_t ws_size,
                              hipStream_t stream) {
  const int*   dtok = (const int*)d_in[0];    // draft_tokens (B,K)
  const float* dp   = (const float*)d_in[1];  // draft_probs  (B,K,V)
  // d_in[2] = oracle_tokens (unused by the reference math)
  const float* op   = (const float*)d_in[3];  // oracle_probs (B,K+1,V)
  const float* uni  = (const float*)d_in[4];  // uniforms     (B,K)
  const float* su   = (const float*)d_in[5];  // sample_u     (B,)
  const int B = in_sizes[5];
  const int K = in_sizes[0] / B;
  const int V = in_sizes[1] / in_sizes[0];
  const int nchunk = V / CHUNK;               // V=128000 -> 500 (divisible)

  float* ws_cs = (float*)d_ws;                // B*nchunk floats (~128 KB)
  int*   out   = (int*)d_out;                 // tokens (B,K+1) ++ num_accepted (B)

  dim3 g1(B, SPLIT);
  rs_chunk_sums<<<g1, TPB1, 0, stream>>>(dtok, dp, op, uni, ws_cs, B, K, V, nchunk);
  rs_finalize<<<B, TPB2, 0, stream>>>(dtok, dp, op, uni, su, ws_cs, out, B, K, V, nchunk);
  (void)n_in; (void)out_size; (void)ws_size;
}